// DGT_6485400616966
// MI455X (gfx1250) — compile-verified
//
#include <hip/hip_runtime.h>
#include <hip/hip_bf16.h>
#include <math.h>

// DGT forward for MI455X (gfx1250, wave32).
// Hybrid strategy: dense WMMA (f32 16x16x4) for tree levels 0..4 (nodes 0..30,
// computed as a 64x32 tile per block), then per-lane adaptive descent for
// levels 5..9 reading x from LDS. Outputs: softmax(W_or[:,leaf]) and
// clip(action_stds[:,leaf], -20, 2).

typedef __attribute__((ext_vector_type(2))) float v2f;
typedef __attribute__((ext_vector_type(8))) float v8f;

#define SAMPLES_PER_BLOCK 64
#define XS 257   // LDS row stride for x tile (odd -> bank-conflict-free columns)
#define PZS 33   // LDS row stride for pred_z tile

__global__ __launch_bounds__(128) void dgt_forward_kernel(
    const float* __restrict__ x,            // [65536, 256]
    const float* __restrict__ W_pred,       // [1023, 256]
    const float* __restrict__ b_pred,       // [1023]
    const float* __restrict__ W_or,         // [16, 1024]
    const float* __restrict__ action_stds,  // [16, 1024]
    float* __restrict__ out)                // [65536*16] out ++ [65536*16] std
{
    extern __shared__ float smem[];
    float* xs = smem;                        // [64][XS]
    float* pz = smem + SAMPLES_PER_BLOCK * XS; // [64][PZS]

    const int tid  = threadIdx.x;
    const int lane = tid & 31;
    const int wv   = tid >> 5;
    const int half = lane >> 4;   // 0: lanes 0-15, 1: lanes 16-31
    const int lm   = lane & 15;

    // ---------------- Phase 0: stage x tile into LDS (coalesced float4) -----
    {
        const float4* xg = (const float4*)(x + (size_t)blockIdx.x * SAMPLES_PER_BLOCK * 256);
        #pragma unroll
        for (int j = tid; j < SAMPLES_PER_BLOCK * 64; j += 128) {
            float4 v = xg[j];
            int r  = j >> 6;          // sample row in tile
            int c  = (j & 63) << 2;   // column (float index)
            float* dst = xs + r * XS + c;
            dst[0] = v.x; dst[1] = v.y; dst[2] = v.z; dst[3] = v.w;
        }
    }
    __syncthreads();

    // ---------------- Phase 1: dense WMMA, nodes 0..31, f32 16x16x4 ---------
    // Wave w owns samples [16w, 16w+16). Two N-tiles: nodes [0,16) and [16,32).
    // A (16x4, f32): lane holds row M=lm, K = 2*half + v   (v = vgpr 0/1)
    // B (4x16, f32): lane holds col N=lm, K = 2*half + v ; B[k][n] = W_pred[n][k]
    // C/D (16x16, f32): lane/vgpr v holds M = v + 8*half, N = lm
    {
        v8f acc0 = {}; v8f acc1 = {};
        const int arow  = (wv * 16 + lm) * XS + 2 * half;
        const int brow0 = lm * 256 + 2 * half;          // W_pred row lm
        const int brow1 = (lm + 16) * 256 + 2 * half;   // W_pred row lm+16
        #pragma unroll 4
        for (int t = 0; t < 64; ++t) {
            const int k0 = t * 4;
            v2f a, b0, b1;
            a[0]  = xs[arow + k0];        a[1]  = xs[arow + k0 + 1];
            b0[0] = W_pred[brow0 + k0];   b0[1] = W_pred[brow0 + k0 + 1];
            b1[0] = W_pred[brow1 + k0];   b1[1] = W_pred[brow1 + k0 + 1];
            acc0 = __builtin_amdgcn_wmma_f32_16x16x4_f32(
                false, a, false, b0, (short)0, acc0, false, false);
            acc1 = __builtin_amdgcn_wmma_f32_16x16x4_f32(
                false, a, false, b1, (short)0, acc1, false, false);
        }
        const float bb0 = b_pred[lm];
        const float bb1 = b_pred[16 + lm];
        #pragma unroll
        for (int v = 0; v < 8; ++v) {
            const int m = wv * 16 + v + 8 * half;       // sample row in tile
            pz[m * PZS + lm]      = acc0[v] + bb0;      // node = lm
            pz[m * PZS + 16 + lm] = acc1[v] + bb1;      // node = 16+lm
        }
    }
    __syncthreads();

    // ---------------- Phase 2: per-sample descent (waves 0,1; 1 lane = 1 sample)
    if (wv < 2) {
        const int s = wv * 32 + lane;                   // 0..63 in tile
        int node = 0;
        #pragma unroll
        for (int lvl = 0; lvl < 5; ++lvl) {
            float z = pz[s * PZS + node];
            node = 2 * node + 1 + (z < 0.0f ? 1 : 0);   // z>=0 -> left
        }
        // Adaptive levels 5..9: node in [31,62] .. [511,1022]
        const float* xrow = xs + s * XS;
        #pragma unroll
        for (int lvl = 5; lvl < 10; ++lvl) {
            const float4* wr = (const float4*)(W_pred + node * 256);
            float a0 = 0.f, a1 = 0.f, a2 = 0.f, a3 = 0.f;
            #pragma unroll 8
            for (int k = 0; k < 64; ++k) {
                float4 w4 = wr[k];
                a0 = fmaf(xrow[4 * k],     w4.x, a0);
                a1 = fmaf(xrow[4 * k + 1], w4.y, a1);
                a2 = fmaf(xrow[4 * k + 2], w4.z, a2);
                a3 = fmaf(xrow[4 * k + 3], w4.w, a3);
            }
            float z = b_pred[node] + ((a0 + a1) + (a2 + a3));
            node = 2 * node + 1 + (z < 0.0f ? 1 : 0);
        }
        const int leaf = node - 1023;                   // [0, 1023]

        // ---------------- Phase 3: outputs ---------------------------------
        float vals[16];
        float m = -1e30f;
        #pragma unroll
        for (int o = 0; o < 16; ++o) {
            vals[o] = W_or[o * 1024 + leaf];
            m = fmaxf(m, vals[o]);
        }
        float ssum = 0.f;
        #pragma unroll
        for (int o = 0; o < 16; ++o) {
            vals[o] = __expf(vals[o] - m);
            ssum += vals[o];
        }
        const float inv = 1.0f / ssum;

        const size_t gs = (size_t)blockIdx.x * SAMPLES_PER_BLOCK + s;
        float4* out4 = (float4*)(out + gs * 16);
        float4* std4 = (float4*)(out + (size_t)65536 * 16 + gs * 16);
        #pragma unroll
        for (int q = 0; q < 4; ++q) {
            float4 ov, sv;
            ov.x = vals[4 * q]     * inv;
            ov.y = vals[4 * q + 1] * inv;
            ov.z = vals[4 * q + 2] * inv;
            ov.w = vals[4 * q + 3] * inv;
            sv.x = fminf(2.0f, fmaxf(-20.0f, action_stds[(4 * q)     * 1024 + leaf]));
            sv.y = fminf(2.0f, fmaxf(-20.0f, action_stds[(4 * q + 1) * 1024 + leaf]));
            sv.z = fminf(2.0f, fmaxf(-20.0f, action_stds[(4 * q + 2) * 1024 + leaf]));
            sv.w = fminf(2.0f, fmaxf(-20.0f, action_stds[(4 * q + 3) * 1024 + leaf]));
            out4[q] = ov;
            std4[q] = sv;
        }
    }
}

extern "C" void kernel_launch(void* const* d_in, const int* in_sizes, int n_in,
                              void* d_out, int out_size, void* d_ws, size_t ws_size,
                              hipStream_t stream) {
    const float* x           = (const float*)d_in[0];
    const float* W_pred      = (const float*)d_in[1];
    const float* b_pred      = (const float*)d_in[2];
    const float* W_or        = (const float*)d_in[3];
    const float* action_stds = (const float*)d_in[4];
    float* out = (float*)d_out;

    const size_t shmem = (size_t)(SAMPLES_PER_BLOCK * XS + SAMPLES_PER_BLOCK * PZS) * sizeof(float);
    (void)hipFuncSetAttribute((const void*)dgt_forward_kernel,
                              hipFuncAttributeMaxDynamicSharedMemorySize, (int)shmem);

    dim3 grid(65536 / SAMPLES_PER_BLOCK);  // 1024 blocks
    dim3 block(128);                       // 4 wave32 waves
    dgt_forward_kernel<<<grid, block, shmem, stream>>>(
        x, W_pred, b_pred, W_or, action_stds, out);
}